// ConceptLayer_2181843386586
// MI455X (gfx1250) — compile-verified
//
#include <hip/hip_runtime.h>
#include <stdint.h>

typedef __attribute__((ext_vector_type(2))) float v2f;
typedef __attribute__((ext_vector_type(8))) float v8f;
typedef __attribute__((ext_vector_type(4))) unsigned int v4u;
typedef __attribute__((ext_vector_type(8))) int          v8i;
typedef __attribute__((ext_vector_type(4))) int          v4i;

// ---------------- LDS layout (bytes: 242,688 of the 320KB WGP LDS) ---------
// [0 .. WREG)            : weight staging region (We in GEMM1, Wd in GEMM2)
// [WREG .. WREG+LOGITS)  : 128 x 66 f32 logits / concepts scratch
#define WE_PITCH   772                  // 772 % 64 == 4 -> conflict-free b64
#define WD_PITCH   68                   // 68  % 64 == 4 -> conflict-free b64
#define LOG_PITCH  66
#define WREG_FLTS  (768 * WD_PITCH)     // 52,224 floats (>= 64*772 = 49,408)
#define LOG_OFF    WREG_FLTS
#define SMEM_FLTS  (WREG_FLTS + 128 * LOG_PITCH)

// Tensor Data Mover: 2D strided copy global->LDS using iterate mode so the
// LDS row pitch is decoupled from the tensor row stride (bank-conflict pad).
// D# packing per CDNA5 ISA ch.8 (group0/1 + group2 iterate fields).
// This toolchain exposes the 6-arg builtin: (g0, g1, g2, g3, g4, cpol).
__device__ static inline void tdm_load_2d(uint32_t lds_off_bytes,
                                          const void* gsrc,
                                          uint32_t row_elems,   // f32 per row
                                          uint32_t nrows,       // <= 256
                                          uint32_t g_stride_el, // global row stride
                                          uint32_t l_pitch_el)  // LDS row pitch
{
    const uint64_t ga = (uint64_t)(uintptr_t)gsrc;
    v4u g0;
    g0.x = 1u;                                    // count=1, user descriptor
    g0.y = lds_off_bytes;                         // lds_addr [63:32]
    g0.z = (uint32_t)ga;                          // global_addr lo
    g0.w = (uint32_t)(ga >> 32) | (2u << 30);     // global_addr hi | type=2
    v8i g1;
    g1[0] = (int)((2u << 16) | (1u << 19));       // data_size=4B, iterate_enable
    g1[1] = (int)((row_elems & 0xFFFFu) << 16);   // tensor_dim0 lo16 @ [63:48]
    g1[2] = (int)((row_elems >> 16) | (1u << 16));// tensor_dim0 hi16 | tensor_dim1=1
    g1[3] = (int)((row_elems & 0xFFFFu) << 16);   // tile_dim0 @ [127:112]
    g1[4] = 1;                                    // tile_dim1=1, tile_dim2=0
    g1[5] = (int)row_elems;                       // tensor_dim0_stride lo32
    g1[6] = 0;                                    // stride hi | dim1_stride lo
    g1[7] = 0;
    v4i g2;
    g2[0] = 1;                                    // tensor_dim2 (unused, iterate)
    g2[1] = (int)l_pitch_el;                      // lds_addr_increment / iter
    g2[2] = (int)g_stride_el;                     // global_addr_increment lo
    g2[3] = (int)((nrows - 1u) << 16);            // iterate_count @ [127:112]
    v4i g3 = {0, 0, 0, 0};
    v8i g4 = {0, 0, 0, 0, 0, 0, 0, 0};            // 6-arg form: extra group, zeroed
    __builtin_amdgcn_tensor_load_to_lds(g0, g1, g2, g3, g4, 0);
}

__global__ __launch_bounds__(256)
void concept_layer_fused(const float* __restrict__ h,
                         const float* __restrict__ We,
                         const float* __restrict__ be,
                         const float* __restrict__ Wd,
                         const float* __restrict__ bd,
                         float* __restrict__ h_out,
                         float* __restrict__ concepts_out)
{
    constexpr int D = 768, M = 64, ROWS = 128;
    __shared__ float smem[SMEM_FLTS];

    const int tid   = threadIdx.x;
    const int wave  = tid >> 5;            // 0..7 (wave32)
    const int lane  = tid & 31;
    const int l16   = lane & 15;
    const int lhalf = lane >> 4;           // frag half: K={0,1} vs K={2,3}
    const size_t wg_row0 = (size_t)blockIdx.x * ROWS;

    const uint32_t smem_base =
        (uint32_t)(uintptr_t)(__attribute__((address_space(3))) float*)smem;

    // ---- TDM: stage We [64][768] into LDS at pitch 772 (wave 0 only) ------
    if (wave == 0) {
        tdm_load_2d(smem_base, We, D, M, D, WE_PITCH);
        __builtin_amdgcn_s_wait_tensorcnt(0);
    }
    __syncthreads();

    // ============ Phase A: logits = h @ We^T + be (f32 WMMA) ================
    {
        const int row0 = wave * 16;
        const float* __restrict__ hrow = h + (wg_row0 + row0 + l16) * D;
        v8f acc[4] = {};
        for (int k0 = 0; k0 < D; k0 += 64) {
            if (k0 + 256 < D) __builtin_prefetch(hrow + k0 + 256, 0, 3);
            #pragma unroll
            for (int ki = 0; ki < 16; ++ki) {
                const int kk = k0 + ki * 4 + 2 * lhalf;
                v2f a; a.x = hrow[kk]; a.y = hrow[kk + 1];   // A: h streaming
                #pragma unroll
                for (int t = 0; t < 4; ++t) {                // B: We from LDS
                    const int o = (t * 16 + l16) * WE_PITCH + kk;
                    v2f b; b.x = smem[o]; b.y = smem[o + 1];
                    acc[t] = __builtin_amdgcn_wmma_f32_16x16x4_f32(
                        false, a, false, b, (short)0, acc[t], false, false);
                }
            }
        }
        #pragma unroll
        for (int t = 0; t < 4; ++t) {
            const int n = t * 16 + l16;
            const float bias = be[n];
            #pragma unroll
            for (int r = 0; r < 8; ++r) {
                const int mrow = r + 8 * lhalf;       // C layout: lanes16-31=M+8
                smem[LOG_OFF + (row0 + mrow) * LOG_PITCH + n] = acc[t][r] + bias;
            }
        }
    }
    __syncthreads();

    // ---- TDM: stage Wd [768][64] into LDS at pitch 68 (3 x 256 rows), -----
    // ---- asynchronous: overlaps with the Phase B sort below. ---------------
    if (wave == 0) {
        #pragma unroll
        for (int j = 0; j < 3; ++j)
            tdm_load_2d(smem_base + (uint32_t)j * 256u * WD_PITCH * 4u,
                        Wd + (size_t)j * 256 * M, M, 256, M, WD_PITCH);
    }

    // ============ Phase B: sparsemax + top-8, one thread per row ============
    if (tid < ROWS) {
        float* __restrict__ Lrow = &smem[LOG_OFF + tid * LOG_PITCH];
        float zs[64];
        float zmax = -__builtin_inff();
        #pragma unroll
        for (int i = 0; i < 64; ++i) { zs[i] = Lrow[i]; zmax = fmaxf(zmax, zs[i]); }
        #pragma unroll
        for (int i = 0; i < 64; ++i) zs[i] -= zmax;

        // fully-unrolled descending bitonic sort (constant indices -> VALU)
        #pragma unroll
        for (int k = 2; k <= 64; k <<= 1) {
            #pragma unroll
            for (int j = k >> 1; j > 0; j >>= 1) {
                #pragma unroll
                for (int i = 0; i < 64; ++i) {
                    const int l = i ^ j;
                    if (l > i) {
                        const bool dsc = ((i & k) == 0);
                        const float x = zs[i], y = zs[l];
                        const float hi = fmaxf(x, y), lo = fminf(x, y);
                        zs[i] = dsc ? hi : lo;
                        zs[l] = dsc ? lo : hi;
                    }
                }
            }
        }
        float cs = 0.f, csk = 0.f; int ksup = 1;
        #pragma unroll
        for (int i = 0; i < 64; ++i) {
            cs += zs[i];
            const float thr = (cs - 1.0f) / (float)(i + 1);
            if (zs[i] > thr) { ksup = i + 1; csk = cs; }
        }
        const float tau = (csk - 1.0f) / (float)ksup;
        float psum = 0.f;
        #pragma unroll
        for (int i = 0; i < 64; ++i) psum += fmaxf(Lrow[i] - zmax - tau, 0.f);
        const float inv  = (psum > 0.f) ? (1.0f / psum) : 0.f;
        const float v8th = fmaxf(zs[7] - tau, 0.f) * inv;   // 8th-largest prob
        #pragma unroll
        for (int i = 0; i < 64; ++i) {
            float p = fmaxf(Lrow[i] - zmax - tau, 0.f) * inv;
            p = (p > 0.f && p >= v8th) ? p : 0.f;           // top-8 keep
            Lrow[i] = p;
        }
    }
    if (wave == 0) __builtin_amdgcn_s_wait_tensorcnt(0);    // Wd staged
    __syncthreads();

    // ================= emit concepts, coalesced from LDS ====================
    {
        float* __restrict__ cptr = concepts_out + wg_row0 * M;
        for (int f = tid; f < ROWS * M; f += 256)
            cptr[f] = smem[LOG_OFF + (f >> 6) * LOG_PITCH + (f & 63)];
    }

    // ============ Phase C: h_rec = concepts @ Wd^T + bd (all-LDS) ===========
    {
        const int rbase = wave * 16;
        const float* __restrict__ arow = &smem[LOG_OFF + (rbase + l16) * LOG_PITCH];
        #pragma unroll 1
        for (int ch = 0; ch < 6; ++ch) {       // 6 chunks x 8 col-tiles = 768
            v8f acc[8] = {};
            #pragma unroll 4
            for (int k0 = 0; k0 < M; k0 += 4) {
                const int kk = k0 + 2 * lhalf;
                v2f a; a.x = arow[kk]; a.y = arow[kk + 1];   // A: concepts (LDS)
                #pragma unroll
                for (int t = 0; t < 8; ++t) {                // B: Wd (LDS)
                    const int o = ((ch * 8 + t) * 16 + l16) * WD_PITCH + kk;
                    v2f b; b.x = smem[o]; b.y = smem[o + 1];
                    acc[t] = __builtin_amdgcn_wmma_f32_16x16x4_f32(
                        false, a, false, b, (short)0, acc[t], false, false);
                }
            }
            #pragma unroll
            for (int t = 0; t < 8; ++t) {
                const int n = (ch * 8 + t) * 16 + l16;
                const float bias = bd[n];
                #pragma unroll
                for (int r = 0; r < 8; ++r) {
                    const int mrow = r + 8 * lhalf;
                    h_out[(wg_row0 + rbase + mrow) * D + n] = acc[t][r] + bias;
                }
            }
        }
    }
}

extern "C" void kernel_launch(void* const* d_in, const int* in_sizes, int n_in,
                              void* d_out, int out_size, void* d_ws, size_t ws_size,
                              hipStream_t stream)
{
    const float* h  = (const float*)d_in[0];   // [16,4096,768]
    const float* We = (const float*)d_in[1];   // [64,768]
    const float* be = (const float*)d_in[2];   // [64]
    const float* Wd = (const float*)d_in[3];   // [768,64]
    const float* bd = (const float*)d_in[4];   // [768]

    float* h_out = (float*)d_out;                       // output 0, flat
    const size_t nrows = (size_t)in_sizes[0] / 768;     // B*S = 65536
    float* concepts = h_out + nrows * 768;              // output 1, flat

    const int grid = (int)(nrows / 128);                // 512 workgroups
    hipLaunchKernelGGL(concept_layer_fused, dim3(grid), dim3(256), 0, stream,
                       h, We, be, Wd, bd, h_out, concepts);

    (void)n_in; (void)out_size; (void)d_ws; (void)ws_size;
}